// MatQuatCapNet_70617852280855
// MI455X (gfx1250) — compile-verified
//
#include <hip/hip_runtime.h>
#include <math.h>

// ---------------------------------------------------------------------------
// CDNA5 (gfx1250) implementation of MatQuatCapNet forward.
// Wave32, WMMA f32 16x16x4 for all GEMM-shaped work (implicit-GEMM convs and
// quaternion vote contractions: K=4 matches V_WMMA_F32_16X16X4_F32 exactly).
// EM routing: one workgroup per spatial position, single-pass moments with
// async global->LDS double-buffered staging (ASYNCcnt), LDS atomics.
// ---------------------------------------------------------------------------

typedef __attribute__((ext_vector_type(2))) float v2f;
typedef __attribute__((ext_vector_type(8))) float v8f;

__device__ __forceinline__ v8f wmma4(v2f a, v2f b, v8f c) {
  // D = A(16x4) * B(4x16) + C(16x16), all f32, one instruction (K fully consumed)
  return __builtin_amdgcn_wmma_f32_16x16x4_f32(
      /*neg_a=*/false, a, /*neg_b=*/false, b,
      /*c_mod=*/(short)0, c, /*reuse_a=*/false, /*reuse_b=*/false);
}

#define BN_EPS 1e-5f
#define EPS 1e-10f

// ---------------------------------------------------------------------------
// Per-channel training-mode batchnorm statistics: stats[0..C)=mean, [C..2C)=var
// ---------------------------------------------------------------------------
__global__ __launch_bounds__(256) void bn_stats_kernel(
    const float* __restrict__ x, float* __restrict__ stats,
    int C, int B, int H, int W) {
  int c = blockIdx.x;
  int HW = H * W;
  int N = B * HW;
  float s = 0.f, s2 = 0.f;
  for (int idx = threadIdx.x; idx < N; idx += blockDim.x) {
    int b = idx / HW, r = idx % HW;
    float v = x[((size_t)(b * C + c)) * HW + r];
    s += v; s2 += v * v;
  }
  __shared__ float sh[256], sh2[256];
  sh[threadIdx.x] = s; sh2[threadIdx.x] = s2;
  __syncthreads();
  for (int st = 128; st > 0; st >>= 1) {
    if ((int)threadIdx.x < st) {
      sh[threadIdx.x] += sh[threadIdx.x + st];
      sh2[threadIdx.x] += sh2[threadIdx.x + st];
    }
    __syncthreads();
  }
  if (threadIdx.x == 0) {
    float m = sh[0] / (float)N;
    stats[c] = m;
    stats[C + c] = sh2[0] / (float)N - m * m;  // biased var (jnp default)
  }
}

// ---------------------------------------------------------------------------
// Implicit-GEMM conv (NCHW, OIHW) with BN+ReLU fused into the A-fragment load.
// One wave (32 lanes) per 16x16 output tile; K consumed 4 at a time by WMMA.
// stats==nullptr -> plain ReLU on the input (used for relu(xp)/relu(xa)).
// resid!=nullptr -> residual add before store (preact shortcut).
// M = B*Ho*Wo and Cout are multiples of 16 for every call site.
// ---------------------------------------------------------------------------
__global__ __launch_bounds__(32) void conv_wmma_kernel(
    const float* __restrict__ in, const float* __restrict__ w,
    const float* __restrict__ stats, const float* __restrict__ gamma,
    const float* __restrict__ beta, const float* __restrict__ resid,
    float* __restrict__ out,
    int B, int Cin, int Hi, int Wi, int Cout, int Ho, int Wo,
    int ksz, int stride, int pad) {
  const int lane = threadIdx.x;
  const int mr = lane & 15;
  const int half = lane >> 4;        // lanes 16..31 handle K+2,K+3 of each step
  const int HoWo = Ho * Wo;
  const int m = blockIdx.x * 16 + mr;
  const int b = m / HoWo;
  const int hw = m % HoWo;
  const int ho = hw / Wo, wo = hw % Wo;
  const int ncol = blockIdx.y * 16 + mr;
  const int kk2 = ksz * ksz;
  const int K = Cin * kk2;

  v8f acc = {0.f, 0.f, 0.f, 0.f, 0.f, 0.f, 0.f, 0.f};
  for (int kb = 0; kb < K; kb += 4) {
    const int k0 = kb + half * 2;
    v2f afrag, bfrag;
#pragma unroll
    for (int t = 0; t < 2; ++t) {
      int kk = k0 + t;
      float av = 0.f, bv = 0.f;
      if (kk < K) {
        int ci = kk / kk2, rem = kk % kk2;
        int r = rem / ksz, sx = rem % ksz;
        int hi = ho * stride + r - pad;
        int wi = wo * stride + sx - pad;
        if (hi >= 0 && hi < Hi && wi >= 0 && wi < Wi) {
          float v = in[(((size_t)(b * Cin + ci)) * Hi + hi) * Wi + wi];
          if (stats) {  // fused training-mode BN
            float sc = gamma[ci] * rsqrtf(stats[Cin + ci] + BN_EPS);
            v = sc * (v - stats[ci]) + beta[ci];
          }
          av = fmaxf(v, 0.f);  // ReLU
        }
        bv = w[(size_t)ncol * K + kk];
      }
      afrag[t] = av;
      bfrag[t] = bv;
    }
    acc = wmma4(afrag, bfrag, acc);
  }

#pragma unroll
  for (int r8 = 0; r8 < 8; ++r8) {
    int mo = blockIdx.x * 16 + r8 + half * 8;  // C/D layout: VGPR r8 <-> row
    int b2 = mo / HoWo, hw2 = mo % HoWo;
    int ho2 = hw2 / Wo, wo2 = hw2 % Wo;
    size_t oidx = (((size_t)(b2 * Cout + ncol)) * Ho + ho2) * Wo + wo2;
    float v = acc[r8];
    if (resid) v += resid[oidx];
    out[oidx] = v;
  }
}

// ---------------------------------------------------------------------------
// Rotation matrices R(q*)L(q) packed directly into WMMA B-fragment layout:
// rotB[pair][k][o*4+c] = R_{pair,o}[c][k]   (row stride Npad, zero padded)
// ---------------------------------------------------------------------------
__global__ __launch_bounds__(256) void rotmat_kernel(
    const float* __restrict__ theta, const float* __restrict__ what,
    float* __restrict__ rotB, int npairs, int O, int Npad) {
  int idx = blockIdx.x * blockDim.x + threadIdx.x;
  if (idx >= npairs * O) return;
  int pair = idx / O, o = idx % O;
  float th = theta[idx];
  float wx = what[(size_t)idx * 3 + 0];
  float wy = what[(size_t)idx * 3 + 1];
  float wz = what[(size_t)idx * 3 + 2];
  float inv = rsqrtf(wx * wx + wy * wy + wz * wz);
  float s = sinf(th) + EPS;
  float q0 = cosf(th) + EPS;
  float q1 = s * wx * inv, q2 = s * wy * inv, q3 = s * wz * inv;
  float Wm[4][4] = {
      { q0, -q1, -q2, -q3},
      { q1,  q0, -q3,  q2},
      { q2,  q3,  q0, -q1},
      { q3, -q2,  q1,  q0}};
  const float L2Rm[4][4] = {{1, 1, 1, 1}, {1, 1, -1, -1}, {1, -1, 1, -1}, {1, -1, -1, 1}};
  float* dst = rotB + (size_t)pair * 4 * Npad;
#pragma unroll
  for (int r = 0; r < 4; ++r) {
#pragma unroll
    for (int c = 0; c < 4; ++c) {
      float a = 0.f;
#pragma unroll
      for (int k = 0; k < 4; ++k)  // Wc[r][k]*Wm[k][c], Wc = L2R .* Wm^T
        a += L2Rm[r][k] * Wm[k][r] * Wm[k][c];
      dst[c * Npad + o * 4 + r] = a;  // B[k=c][n=o*4+r] = R[r][c]
    }
  }
  if (o == 0) {  // zero padding columns (fc layer: N=20 padded to 32)
    for (int k = 0; k < 4; ++k)
      for (int n = 4 * O; n < Npad; ++n) dst[k * Npad + n] = 0.f;
  }
}

// ---------------------------------------------------------------------------
// Votes: per (kl,i) pair, GEMM (positions x 4) @ (4 x 4O) — a single
// v_wmma_f32_16x16x4_f32 per 16x16 tile. V[pos][kl*I+i][o][c].
// ---------------------------------------------------------------------------
__global__ __launch_bounds__(32) void votes_wmma_kernel(
    const float* __restrict__ xq, const float* __restrict__ rotB,
    float* __restrict__ V,
    int B, int H, int W, int I, int ksz, int oh, int ow, int O, int Npad) {
  const int lane = threadIdx.x, mr = lane & 15, half = lane >> 4;
  const int pair = blockIdx.x;
  const int npairs = ksz * ksz * I;
  const int kl = pair / I, i = pair % I;
  const int ki = kl / ksz, kj = kl % ksz;
  const int ohow = oh * ow;
  const int m = blockIdx.y * 16 + mr;
  const int b = m / ohow, hw = m % ohow;
  const int h = hw / ow, wc = hw % ow;

  // A fragment: quaternion components (contiguous float2 at K offset)
  const float2 a2 = *(const float2*)(xq +
      ((((size_t)(b * H + h + ki)) * W + (wc + kj)) * I + i) * 4 + half * 2);
  v2f afrag; afrag[0] = a2.x; afrag[1] = a2.y;

  const int N = 4 * O;
  const float* rb = rotB + (size_t)pair * 4 * Npad;
  const int ntiles = Npad / 16;
  for (int nt = 0; nt < ntiles; ++nt) {
    int ncol = nt * 16 + mr;
    v2f bfrag;
    bfrag[0] = rb[(half * 2 + 0) * Npad + ncol];
    bfrag[1] = rb[(half * 2 + 1) * Npad + ncol];
    v8f zero = {0.f, 0.f, 0.f, 0.f, 0.f, 0.f, 0.f, 0.f};
    v8f d = wmma4(afrag, bfrag, zero);
    if (ncol < N) {
#pragma unroll
      for (int r8 = 0; r8 < 8; ++r8) {
        int mo = blockIdx.y * 16 + r8 + half * 8;
        V[((size_t)mo * npairs + pair) * N + ncol] = d[r8];
      }
    }
  }
}

// ---------------------------------------------------------------------------
// EM routing: one workgroup per output position. R (<=800x16) + activations
// in LDS; single-pass moments (Rs, sum rw*V, sum rw*V^2) so the vote tensor
// is streamed 3x instead of 5x. The moments pass double-buffers each wave's
// 512B chunk through LDS via GLOBAL_LOAD_ASYNC_TO_LDS_B128 (ASYNCcnt),
// overlapping fetch with the LDS-atomic reduction. Per-wave consumption only,
// so no cross-wave barrier is needed (async loads complete in order).
// ---------------------------------------------------------------------------
#define EM_MAX_I 800
#define EM_MAX_O 16
#define EM_BS 256

__global__ __launch_bounds__(EM_BS) void em_kernel(
    const float* __restrict__ V, const float* __restrict__ amap,
    const float* __restrict__ beta_u, const float* __restrict__ beta_a,
    float* __restrict__ pose_out, float* __restrict__ act_out,
    int B, int H, int W, int I0, int ksz, int oh, int ow, int O, int n_iters) {
  __shared__ float sR[EM_MAX_I * EM_MAX_O];
  __shared__ float sA[EM_MAX_I];
  __shared__ float sMu[EM_MAX_O * 4];    // accumulates sum(rw*V), then mu
  __shared__ float sSig[EM_MAX_O * 4];   // accumulates sum(rw*V^2), then sigma^2
  __shared__ float sRs[EM_MAX_O];
  __shared__ float sAo[EM_MAX_O];
  __shared__ float sStage[2 * EM_BS * 4];  // async staging: 2 x 256 x float4 = 8KB

  const int tid = threadIdx.x, bs = blockDim.x;
  const int Iem = ksz * ksz * I0;
  const int NP = Iem * O;
  const int pos = blockIdx.x;
  const int ohow = oh * ow;
  const int b = pos / ohow, hw = pos % ohow;
  const int h = hw / ow, wv = hw % ow;

  for (int j = tid; j < Iem; j += bs) {
    int kl = j / I0, i = j % I0;
    int ki = kl / ksz, kj = kl % ksz;
    sA[j] = amap[(((size_t)(b * H + h + ki)) * W + (wv + kj)) * I0 + i];
  }
  const float invO = 1.0f / (float)O;
  for (int p = tid; p < NP; p += bs) sR[p] = invO;
  __syncthreads();

  const float* Vp = V + (size_t)pos * NP * 4;
  // LDS byte offset of this thread's staging slot (low 32 bits of the flat address)
  const unsigned ldsSlot =
      (unsigned)(unsigned long long)(&sStage[(size_t)tid * 4]);
  const unsigned ldsBufStride = (unsigned)(EM_BS * 4 * sizeof(float));
  const int nchunks = (NP + bs - 1) / bs;

  for (int it = 0; it < n_iters; ++it) {
    if (tid < O) sRs[tid] = 0.f;
    if (tid < O * 4) { sMu[tid] = 0.f; sSig[tid] = 0.f; }
    __syncthreads();

    // ---- M-step single pass: Rs, sum(rw*V), sum(rw*V^2) via async staging ----
    {
      // issue chunk 0 into buffer 0
      {
        int p0 = tid; if (p0 >= NP) p0 = NP - 1;  // clamped tail (never consumed)
        unsigned long long ga = (unsigned long long)(Vp + (size_t)p0 * 4);
        asm volatile("global_load_async_to_lds_b128 %0, %1, off"
                     :: "v"(ldsSlot), "v"(ga) : "memory");
      }
      for (int c = 0; c < nchunks; ++c) {
        if (c + 1 < nchunks) {
          int pn = (c + 1) * bs + tid; if (pn >= NP) pn = NP - 1;
          unsigned lds = ldsSlot + (unsigned)((c + 1) & 1) * ldsBufStride;
          unsigned long long ga = (unsigned long long)(Vp + (size_t)pn * 4);
          asm volatile("global_load_async_to_lds_b128 %0, %1, off"
                       :: "v"(lds), "v"(ga) : "memory");
          asm volatile("s_wait_asynccnt 0x1" ::: "memory");
        } else {
          asm volatile("s_wait_asynccnt 0x0" ::: "memory");
        }
        int p = c * bs + tid;
        if (p < NP) {
          int j = p / O, o = p - j * O;
          float rw = sR[p] * sA[j];
          int sb = ((c & 1) * EM_BS + tid) * 4;
          float vx = sStage[sb + 0], vy = sStage[sb + 1];
          float vz = sStage[sb + 2], vw = sStage[sb + 3];
          atomicAdd(&sRs[o], rw);
          atomicAdd(&sMu[o * 4 + 0], rw * vx);
          atomicAdd(&sMu[o * 4 + 1], rw * vy);
          atomicAdd(&sMu[o * 4 + 2], rw * vz);
          atomicAdd(&sMu[o * 4 + 3], rw * vw);
          atomicAdd(&sSig[o * 4 + 0], rw * vx * vx);
          atomicAdd(&sSig[o * 4 + 1], rw * vy * vy);
          atomicAdd(&sSig[o * 4 + 2], rw * vz * vz);
          atomicAdd(&sSig[o * 4 + 3], rw * vw * vw);
        }
      }
    }
    __syncthreads();
    if (tid < O) {
      float rs = sRs[tid] + EPS;
      sRs[tid] = rs;
      float inv = 1.0f / rs;
      float cost = 0.f, bu = beta_u[tid];
      for (int d = 0; d < 4; ++d) {
        float mu = sMu[tid * 4 + d] * inv;
        sMu[tid * 4 + d] = mu;
        float s2 = sSig[tid * 4 + d] * inv - mu * mu + EPS;  // E[V^2]-mu^2
        sSig[tid * 4 + d] = s2;
        cost += (bu + 0.5f * logf(s2)) * rs;
      }
      float lam = 0.01f * (float)(it + 1);
      sAo[tid] = 1.0f / (1.0f + expf(-lam * (beta_a[tid] - cost)));
    }
    __syncthreads();

    // ---- E-step ----
    if (it < n_iters - 1) {
      for (int p = tid; p < NP; p += bs) {
        int j = p / O, o = p - j * O;
        float4 v4 = *(const float4*)(Vp + (size_t)p * 4);
        if (p + bs < NP) __builtin_prefetch(Vp + (size_t)(p + bs) * 4, 0, 0);
        float vv[4] = {v4.x, v4.y, v4.z, v4.w};
        float lnp = 0.f;
        for (int d = 0; d < 4; ++d) {
          float dv = vv[d] - sMu[o * 4 + d];
          float s2 = sSig[o * 4 + d];
          lnp += dv * dv / s2 + logf(6.2831853071795864f * s2);
        }
        sR[p] = logf(sAo[o] + EPS) - 0.5f * lnp;  // logits
      }
      __syncthreads();
      for (int j = tid; j < Iem; j += bs) {  // softmax over O per input cap
        float mx = -3.0e38f;
        for (int o = 0; o < O; ++o) mx = fmaxf(mx, sR[j * O + o]);
        float sum = 0.f;
        for (int o = 0; o < O; ++o) {
          float e = expf(sR[j * O + o] - mx);
          sR[j * O + o] = e; sum += e;
        }
        float inv = 1.0f / sum;
        for (int o = 0; o < O; ++o) sR[j * O + o] *= inv;
      }
      __syncthreads();
    }
  }

  if (tid < O) {
    act_out[(size_t)pos * O + tid] = sAo[tid];
    for (int d = 0; d < 4; ++d)
      pose_out[((size_t)pos * O + tid) * 4 + d] = sMu[tid * 4 + d];
  }
}

// ---------------------------------------------------------------------------
// Primary caps assembly: BN -> pure-quaternion poses (B,16,16,32,4) and
// BN+sigmoid activations (B,16,16,32).
// ---------------------------------------------------------------------------
__global__ __launch_bounds__(256) void build_xq_kernel(
    const float* __restrict__ raw, const float* __restrict__ stats,
    const float* __restrict__ g, const float* __restrict__ bb,
    float* __restrict__ xq, int B) {
  int idx = blockIdx.x * blockDim.x + threadIdx.x;
  if (idx >= B * 16 * 16 * 32) return;
  int cap = idx & 31;
  int w = (idx >> 5) & 15;
  int h = (idx >> 9) & 15;
  int b = idx >> 13;
  float* dst = xq + (size_t)idx * 4;
  dst[0] = 0.f;  // zero scalar part -> pure quaternion
  for (int t = 0; t < 3; ++t) {
    int c = cap * 3 + t;
    float v = raw[(((size_t)(b * 96 + c)) * 16 + h) * 16 + w];
    float sc = g[c] * rsqrtf(stats[96 + c] + BN_EPS);
    dst[1 + t] = sc * (v - stats[c]) + bb[c];
  }
}

__global__ __launch_bounds__(256) void build_a0_kernel(
    const float* __restrict__ raw, const float* __restrict__ stats,
    const float* __restrict__ g, const float* __restrict__ bb,
    float* __restrict__ a0, int B) {
  int idx = blockIdx.x * blockDim.x + threadIdx.x;
  if (idx >= B * 16 * 16 * 32) return;
  int c = idx & 31;
  int w = (idx >> 5) & 15;
  int h = (idx >> 9) & 15;
  int b = idx >> 13;
  float v = raw[(((size_t)(b * 32 + c)) * 16 + h) * 16 + w];
  float sc = g[c] * rsqrtf(stats[32 + c] + BN_EPS);
  v = sc * (v - stats[c]) + bb[c];
  a0[idx] = 1.0f / (1.0f + expf(-v));
}

// ---------------------------------------------------------------------------
// Host-side orchestration
// ---------------------------------------------------------------------------
enum {
  IN_X = 0,
  RB1P_BN1_G, RB1P_BN1_B, RB1P_CONV1_W, RB1P_BN2_G, RB1P_BN2_B, RB1P_CONV2_W, RB1P_SC_W,
  RB2P_BN1_G, RB2P_BN1_B, RB2P_CONV1_W, RB2P_BN2_G, RB2P_BN2_B, RB2P_CONV2_W, RB2P_SC_W,
  RB1A_BN1_G, RB1A_BN1_B, RB1A_CONV1_W, RB1A_BN2_G, RB1A_BN2_B, RB1A_CONV2_W, RB1A_SC_W,
  PRIM_POSE_W, PRIM_POSE_BN_G, PRIM_POSE_BN_B,
  PRIM_ACT_W, PRIM_ACT_BN_G, PRIM_ACT_BN_B,
  C1_THETA, C1_WHAT, C1_BU, C1_BA,
  C2_THETA, C2_WHAT, C2_BU, C2_BA,
  C3_THETA, C3_WHAT, C3_BU, C3_BA,
  FC_THETA, FC_WHAT, FC_BU, FC_BA
};

static void run_preact(hipStream_t s, const float* in, int B, int Cin, int Hi, int Wi,
                       int Cout, int stride,
                       const float* g1, const float* b1, const float* w1,
                       const float* g2, const float* b2, const float* w2,
                       const float* wsc,
                       float* st1, float* st2, float* ybuf, float* scbuf, float* out) {
  int Ho = Hi / stride, Wo = Wi / stride;
  int M = B * Ho * Wo;
  bn_stats_kernel<<<Cin, 256, 0, s>>>(in, st1, Cin, B, Hi, Wi);
  // conv1 3x3 stride s pad 1 (BN+ReLU fused on input)
  conv_wmma_kernel<<<dim3(M / 16, Cout / 16), 32, 0, s>>>(
      in, w1, st1, g1, b1, nullptr, ybuf, B, Cin, Hi, Wi, Cout, Ho, Wo, 3, stride, 1);
  // shortcut 1x1 stride s pad 0 (same fused input transform)
  conv_wmma_kernel<<<dim3(M / 16, Cout / 16), 32, 0, s>>>(
      in, wsc, st1, g1, b1, nullptr, scbuf, B, Cin, Hi, Wi, Cout, Ho, Wo, 1, stride, 0);
  bn_stats_kernel<<<Cout, 256, 0, s>>>(ybuf, st2, Cout, B, Ho, Wo);
  // conv2 3x3 stride 1 pad 1 + residual
  conv_wmma_kernel<<<dim3(M / 16, Cout / 16), 32, 0, s>>>(
      ybuf, w2, st2, g2, b2, scbuf, out, B, Cout, Ho, Wo, Cout, Ho, Wo, 3, 1, 1);
}

extern "C" void kernel_launch(void* const* d_in, const int* in_sizes, int n_in,
                              void* d_out, int out_size, void* d_ws, size_t ws_size,
                              hipStream_t stream) {
  (void)in_sizes; (void)n_in; (void)out_size; (void)ws_size;
  auto F = [&](int i) { return (const float*)d_in[i]; };
  char* ws = (char*)d_ws;
  size_t off = 0;
  auto alloc = [&](size_t nfl) {
    float* p = (float*)(ws + off);
    off += ((nfl * sizeof(float)) + 255) & ~(size_t)255;
    return p;
  };
  const int B = 4;
  const float* x = F(IN_X);

  // workspace layout
  float* stA1 = alloc(2 * 2);
  float* stA2 = alloc(2 * 32);
  float* y1   = alloc((size_t)B * 32 * 32 * 32);
  float* sc1  = alloc((size_t)B * 32 * 32 * 32);
  float* xp1  = alloc((size_t)B * 32 * 32 * 32);
  float* stB1 = alloc(2 * 32);
  float* stB2 = alloc(2 * 64);
  float* y2   = alloc((size_t)B * 64 * 16 * 16);
  float* sc2  = alloc((size_t)B * 64 * 16 * 16);
  float* xp   = alloc((size_t)B * 64 * 16 * 16);
  float* stC1 = alloc(2 * 2);
  float* stC2 = alloc(2 * 32);
  float* y3   = alloc((size_t)B * 32 * 16 * 16);
  float* sc3  = alloc((size_t)B * 32 * 16 * 16);
  float* xa   = alloc((size_t)B * 32 * 16 * 16);
  float* poseR = alloc((size_t)B * 96 * 16 * 16);
  float* stP  = alloc(2 * 96);
  float* xq   = alloc((size_t)B * 16 * 16 * 32 * 4);
  float* actR = alloc((size_t)B * 32 * 16 * 16);
  float* stAc = alloc(2 * 32);
  float* a0   = alloc((size_t)B * 16 * 16 * 32);
  float* rotB1 = alloc((size_t)800 * 4 * 64);
  float* V1   = alloc((size_t)576 * 800 * 64);
  float* p1   = alloc((size_t)B * 12 * 12 * 16 * 4);
  float* a1   = alloc((size_t)B * 12 * 12 * 16);
  float* rotB2 = alloc((size_t)400 * 4 * 64);
  float* V2   = alloc((size_t)256 * 400 * 64);
  float* p2   = alloc((size_t)B * 8 * 8 * 16 * 4);
  float* a2   = alloc((size_t)B * 8 * 8 * 16);
  float* rotB3 = alloc((size_t)400 * 4 * 64);
  float* V3   = alloc((size_t)64 * 400 * 64);
  float* p3   = alloc((size_t)B * 4 * 4 * 16 * 4);
  float* a3   = alloc((size_t)B * 4 * 4 * 16);
  float* rotBf = alloc((size_t)16 * 4 * 32);
  float* Vf   = alloc((size_t)64 * 16 * 20);

  // --- ResNet stem ---
  run_preact(stream, x,   B,  2, 32, 32, 32, 1,
             F(RB1P_BN1_G), F(RB1P_BN1_B), F(RB1P_CONV1_W),
             F(RB1P_BN2_G), F(RB1P_BN2_B), F(RB1P_CONV2_W), F(RB1P_SC_W),
             stA1, stA2, y1, sc1, xp1);
  run_preact(stream, xp1, B, 32, 32, 32, 64, 2,
             F(RB2P_BN1_G), F(RB2P_BN1_B), F(RB2P_CONV1_W),
             F(RB2P_BN2_G), F(RB2P_BN2_B), F(RB2P_CONV2_W), F(RB2P_SC_W),
             stB1, stB2, y2, sc2, xp);
  run_preact(stream, x,   B,  2, 32, 32, 32, 2,
             F(RB1A_BN1_G), F(RB1A_BN1_B), F(RB1A_CONV1_W),
             F(RB1A_BN2_G), F(RB1A_BN2_B), F(RB1A_CONV2_W), F(RB1A_SC_W),
             stC1, stC2, y3, sc3, xa);

  // --- Primary capsules (1x1 convs with ReLU-only input path, then BN) ---
  conv_wmma_kernel<<<dim3(B * 16 * 16 / 16, 96 / 16), 32, 0, stream>>>(
      xp, F(PRIM_POSE_W), nullptr, nullptr, nullptr, nullptr, poseR,
      B, 64, 16, 16, 96, 16, 16, 1, 1, 0);
  bn_stats_kernel<<<96, 256, 0, stream>>>(poseR, stP, 96, B, 16, 16);
  build_xq_kernel<<<(B * 16 * 16 * 32 + 255) / 256, 256, 0, stream>>>(
      poseR, stP, F(PRIM_POSE_BN_G), F(PRIM_POSE_BN_B), xq, B);

  conv_wmma_kernel<<<dim3(B * 16 * 16 / 16, 32 / 16), 32, 0, stream>>>(
      xa, F(PRIM_ACT_W), nullptr, nullptr, nullptr, nullptr, actR,
      B, 32, 16, 16, 32, 16, 16, 1, 1, 0);
  bn_stats_kernel<<<32, 256, 0, stream>>>(actR, stAc, 32, B, 16, 16);
  build_a0_kernel<<<(B * 16 * 16 * 32 + 255) / 256, 256, 0, stream>>>(
      actR, stAc, F(PRIM_ACT_BN_G), F(PRIM_ACT_BN_B), a0, B);

  // --- Conv capsule layers (k=5, s=1, 2 EM iters) ---
  auto run_caps = [&](const float* xin, const float* ain, int H, int I0,
                      const float* th, const float* wh, const float* bu, const float* ba,
                      float* rotB, float* Vb, float* pout, float* aout) {
    const int k = 5, O = 16, oh = H - 4;
    rotmat_kernel<<<(k * k * I0 * O + 255) / 256, 256, 0, stream>>>(
        th, wh, rotB, k * k * I0, O, 64);
    votes_wmma_kernel<<<dim3(k * k * I0, B * oh * oh / 16), 32, 0, stream>>>(
        xin, rotB, Vb, B, H, H, I0, k, oh, oh, O, 64);
    em_kernel<<<B * oh * oh, EM_BS, 0, stream>>>(
        Vb, ain, bu, ba, pout, aout, B, H, H, I0, k, oh, oh, O, 2);
  };
  run_caps(xq, a0, 16, 32, F(C1_THETA), F(C1_WHAT), F(C1_BU), F(C1_BA), rotB1, V1, p1, a1);
  run_caps(p1, a1, 12, 16, F(C2_THETA), F(C2_WHAT), F(C2_BU), F(C2_BA), rotB2, V2, p2, a2);
  run_caps(p2, a2,  8, 16, F(C3_THETA), F(C3_WHAT), F(C3_BU), F(C3_BA), rotB3, V3, p3, a3);

  // --- FC capsules: votes per input-cap (k=1), EM over full 4x4x16 receptive field ---
  float* out = (float*)d_out;
  rotmat_kernel<<<1, 256, 0, stream>>>(F(FC_THETA), F(FC_WHAT), rotBf, 16, 5, 32);
  votes_wmma_kernel<<<dim3(16, B * 4 * 4 / 16), 32, 0, stream>>>(
      p3, rotBf, Vf, B, 4, 4, 16, 1, 4, 4, 5, 32);
  em_kernel<<<B, EM_BS, 0, stream>>>(
      Vf, a3, F(FC_BU), F(FC_BA), out, out + B * 5 * 4, B, 4, 4, 16, 4, 1, 1, 5, 2);
}